// NNG_13529146982773
// MI455X (gfx1250) — compile-verified
//
#include <hip/hip_runtime.h>
#include <math.h>

typedef float v2f __attribute__((ext_vector_type(2)));
typedef float v8f __attribute__((ext_vector_type(8)));

#define N_PTS   18432            // 96 * 192
#define K_OUT   8
#define K_SEL   9                // k_neighbors + 1 (self)
#define TILE    16
#define TSTRIDE 20               // row stride: 80B = 16B-aligned, conflict-free
#define WAVES   8
#define THREADS 256
#define NTILES  (N_PTS / TILE)   // 1152
// each wave handles 2 column tiles per iteration -> 16 tiles per block-iter
#define ITERS   (NTILES / (WAVES * 2))  // 72

// ---------------------------------------------------------------------------
// Kernel 1: lat/lon (degrees) -> unit-sphere coords (padded to float4 w=0)
// plus per-point squared norm (kept separately to mirror the reference math).
// ---------------------------------------------------------------------------
__global__ void NNG_coords_kernel(const float* __restrict__ lat,
                                  const float* __restrict__ lon,
                                  float4* __restrict__ coords4,
                                  float* __restrict__ sq)
{
    int p = blockIdx.x * blockDim.x + threadIdx.x;
    if (p >= N_PTS) return;
    const float DEG2RAD = 0.017453292519943295f;
    float la = lat[p] * DEG2RAD;
    float lo = lon[p] * DEG2RAD;
    float cla = cosf(la), sla = sinf(la);
    float clo = cosf(lo), slo = sinf(lo);
    float x = cla * clo;
    float y = cla * slo;
    float z = sla;
    coords4[p] = make_float4(x, y, z, 0.0f);
    sq[p] = x * x + y * y + z * z;
}

// ---------------------------------------------------------------------------
// Register-resident sorted top-9, ascending by (value, index).
// One rare branch per candidate; inside is a branchless cndmask bubble chain.
// ---------------------------------------------------------------------------
__device__ __forceinline__ void topk9_insert(float v, int idx,
                                             float kv[K_SEL], int ki[K_SEL])
{
    bool better = (v < kv[K_SEL - 1]) |
                  ((v == kv[K_SEL - 1]) & (idx < ki[K_SEL - 1]));
    if (__builtin_expect(better, 0)) {
        kv[K_SEL - 1] = v;
        ki[K_SEL - 1] = idx;
#pragma unroll
        for (int s = K_SEL - 1; s > 0; --s) {
            bool sw = (kv[s] < kv[s - 1]) |
                      ((kv[s] == kv[s - 1]) & (ki[s] < ki[s - 1]));
            float hv = sw ? kv[s - 1] : kv[s];
            float lv = sw ? kv[s]     : kv[s - 1];
            int   hi = sw ? ki[s - 1] : ki[s];
            int   li = sw ? ki[s]     : ki[s - 1];
            kv[s] = hv; kv[s - 1] = lv;
            ki[s] = hi; ki[s - 1] = li;
        }
    }
}

// Wave-private LDS handoff: lanes of a wave32 run in lockstep; ordering of
// the DS ops is guaranteed by DScnt + a wavefront-scope fence (no s_barrier).
__device__ __forceinline__ void wave_lds_fence()
{
    __builtin_amdgcn_fence(__ATOMIC_RELEASE, "wavefront");
    __builtin_amdgcn_wave_barrier();
    __builtin_amdgcn_fence(__ATOMIC_ACQUIRE, "wavefront");
}

// ---------------------------------------------------------------------------
// Kernel 2: one block per 16-row band, 8 independent waves. Each wave per
// iteration computes TWO 16x16 dot tiles via V_WMMA_F32_16X16X4_F32
// (K padded 3->4 with zeros), converts to d2 through a wave-private LDS
// transpose, and streams per-(row, column-subset) top-9 lists in registers:
// lanes 0-15 own tile 0's columns, lanes 16-31 own tile 1's columns.
// Candidates are fetched 4-at-a-time (ds_load_b128); a v_min_f32 reduction
// plus one v_cmpx gives a single rare branch per group (~1.5 instr/cand).
// ---------------------------------------------------------------------------
__launch_bounds__(THREADS)
__global__ void NNG_knn_kernel(const float4* __restrict__ coords4,
                               const float*  __restrict__ sq,
                               float* __restrict__ out_dists,  // 8N floats
                               int*   __restrict__ out_src,    // 9N ints
                               int*   __restrict__ out_dst)    // 9N ints
{
    __shared__ __align__(16) float tileL[WAVES][2][TILE][TSTRIDE];
    __shared__ float candV[WAVES * 2][TILE][K_SEL];     // 16 lists per row
    __shared__ int   candI[WAVES * 2][TILE][K_SEL];

    const int tid  = threadIdx.x;
    const int wave = tid >> 5;
    const int lane = tid & 31;
    const int g    = lane >> 4;     // lane half-group: 0 -> K{0,1}, 1 -> K{2,3}
    const int ln   = lane & 15;
    const int row_base = blockIdx.x * TILE;

    // --- A fragment: 16x4 f32, lane ln = row M, half-group g picks K pair ---
    float4 rc = coords4[row_base + ln];
    v2f a;
    a.x = g ? rc.z : rc.x;
    a.y = g ? rc.w : rc.y;

    // Squared norms for this lane's 8 WMMA output rows: M = r + 8*g
    float sqi[8];
#pragma unroll
    for (int r = 0; r < 8; ++r) sqi[r] = sq[row_base + r + 8 * g];

    // Per-(row, half) streaming top-9: all 32 lanes own a list.
    float kv[K_SEL];
    int   ki[K_SEL];
#pragma unroll
    for (int s = 0; s < K_SEL; ++s) { kv[s] = INFINITY; ki[s] = 0x7fffffff; }

    for (int t = 0; t < ITERS; ++t) {
        // Wave handles column tiles ct and ct+1; stride 16 tiles per iter.
        const int ct        = t * (WAVES * 2) + wave * 2;
        const int col_base0 = ct * TILE;
        const int col_base1 = col_base0 + TILE;

        // --- B fragments: 4x16 f32, lane ln = col N, same K-pair split ---
        float4 c0 = coords4[col_base0 + ln];
        float4 c1 = coords4[col_base1 + ln];
        // Warm caches for next iteration's B tiles (speculative: an
        // out-of-range prefetch past the table is silently dropped).
        __builtin_prefetch(&coords4[col_base0 + WAVES * 2 * TILE + ln], 0, 3);
        __builtin_prefetch(&coords4[col_base1 + WAVES * 2 * TILE + ln], 0, 3);

        v2f b0, b1;
        b0.x = g ? c0.z : c0.x;  b0.y = g ? c0.w : c0.y;
        b1.x = g ? c1.z : c1.x;  b1.y = g ? c1.w : c1.y;

        v8f cz = {0.f, 0.f, 0.f, 0.f, 0.f, 0.f, 0.f, 0.f};
        v8f d0 = __builtin_amdgcn_wmma_f32_16x16x4_f32(
            false, a, false, b0, (short)0, cz, false, false);
        v8f d1 = __builtin_amdgcn_wmma_f32_16x16x4_f32(
            false, a, false, b1, (short)0, cz, false, false);

        // d2 = sq_i + sq_j - 2*dot ; scatter both tiles row-major into LDS
        float sqj0 = sq[col_base0 + ln];
        float sqj1 = sq[col_base1 + ln];
#pragma unroll
        for (int r = 0; r < 8; ++r) {
            tileL[wave][0][r + 8 * g][ln] = sqi[r] + sqj0 - 2.0f * d0[r];
            tileL[wave][1][r + 8 * g][ln] = sqi[r] + sqj1 - 2.0f * d1[r];
        }
        wave_lds_fence();

        // Selection: lane (g, ln) owns row ln over tile-half g's 16 columns.
        const int idx_base = (ct + g) * TILE;
        const float4* rowq = (const float4*)&tileL[wave][g][ln][0];
#pragma unroll
        for (int cq = 0; cq < TILE / 4; ++cq) {
            float4 q = rowq[cq];                     // ds_load_b128
            // "any element <= thr"  ==  "min4(q) <= thr" (no NaNs here):
            // 3x v_min_f32 + 1x v_cmpx, one rare branch per 4 candidates.
            float m = fminf(fminf(q.x, q.y), fminf(q.z, q.w));
            if (__builtin_expect(m <= kv[K_SEL - 1], 0)) {
                topk9_insert(q.x, idx_base + 4 * cq + 0, kv, ki);
                topk9_insert(q.y, idx_base + 4 * cq + 1, kv, ki);
                topk9_insert(q.z, idx_base + 4 * cq + 2, kv, ki);
                topk9_insert(q.w, idx_base + 4 * cq + 3, kv, ki);
            }
        }
        wave_lds_fence();  // reads done before next iteration overwrites
    }

    // --- Publish 16 candidate lists per row, block-wide merge ---
#pragma unroll
    for (int s = 0; s < K_SEL; ++s) {
        candV[wave * 2 + g][ln][s] = kv[s];
        candI[wave * 2 + g][ln][s] = ki[s];
    }
    __syncthreads();

    if (tid < TILE) {
        const int row = tid;
        float fv[K_SEL];
        int   fi[K_SEL];
#pragma unroll
        for (int s = 0; s < K_SEL; ++s) { fv[s] = INFINITY; fi[s] = 0x7fffffff; }

        for (int l = 0; l < WAVES * 2; ++l) {
#pragma unroll
            for (int s = 0; s < K_SEL; ++s) {
                topk9_insert(candV[l][row][s], candI[l][row][s], fv, fi);
            }
        }

        const int i = row_base + row;
        // fv[0] is self (d2 ~ 0) -> dropped, like reference idx[:, 1:]
#pragma unroll
        for (int j = 1; j < K_SEL; ++j) {
            float v = fv[j] > 0.0f ? fv[j] : 0.0f;               // maximum(v,0)
            float dist = (v > 0.0f) ? sqrtf(v) : 0.0f;           // guarded sqrt
            out_dists[(size_t)i * K_OUT + (j - 1)] = dist;
            out_dst  [(size_t)i * K_OUT + (j - 1)] = fi[j];
            out_src  [(size_t)i * K_OUT + (j - 1)] = i;
        }
        // self-loop tail: src/dst append arange(N)
        out_src[(size_t)N_PTS * K_OUT + i] = i;
        out_dst[(size_t)N_PTS * K_OUT + i] = i;
    }
}

// ---------------------------------------------------------------------------
extern "C" void kernel_launch(void* const* d_in, const int* in_sizes, int n_in,
                              void* d_out, int out_size, void* d_ws, size_t ws_size,
                              hipStream_t stream)
{
    (void)in_sizes; (void)n_in; (void)out_size; (void)ws_size;

    const float* lat = (const float*)d_in[0];  // lat_grid, N f32
    const float* lon = (const float*)d_in[1];  // lon_grid, N f32
    // d_in[2] = k_neighbors (==8, baked in as K_OUT)

    // Workspace: coords4 (N * float4) | sq (N * float)  ~= 360 KB
    float4* coords4 = (float4*)d_ws;
    float*  sqn     = (float*)((char*)d_ws + (size_t)N_PTS * sizeof(float4));

    // Output tuple, concatenated flat: dists f32[8N] | src i32[9N] | dst i32[9N]
    float* out_dists = (float*)d_out;
    int*   out_src   = (int*)((float*)d_out + (size_t)8  * N_PTS);
    int*   out_dst   = (int*)((float*)d_out + (size_t)17 * N_PTS);

    NNG_coords_kernel<<<(N_PTS + 255) / 256, 256, 0, stream>>>(lat, lon, coords4, sqn);
    NNG_knn_kernel<<<NTILES, THREADS, 0, stream>>>(coords4, sqn,
                                                   out_dists, out_src, out_dst);
}